// RmstWL_20959440404552
// MI455X (gfx1250) — compile-verified
//
#include <hip/hip_runtime.h>

#define MAX_DEG    8
#define BLOCK_NETS 256
#define INF_F      1e30f

// CDNA5 async global->LDS copy path (gfx1250). Guarded so compile never fails
// if the builtin name differs on this toolchain; fallback is a plain LDS copy.
#if __has_builtin(__builtin_amdgcn_global_load_async_to_lds_b32)
#define USE_ASYNC_LDS 1
#else
#define USE_ASYNC_LDS 0
#endif

typedef __attribute__((address_space(1))) int GInt;   // global (device) int
typedef __attribute__((address_space(3))) int LInt;   // LDS int

__global__ __launch_bounds__(BLOCK_NETS) void rmst_wl_kernel(
    const float* __restrict__ pos,
    const int*   __restrict__ flat_netpin,
    const int*   __restrict__ netpin_start,
    const int*   __restrict__ ignore_net_degree_p,
    float*       __restrict__ out,
    int n_nets, int num_pins)
{
    __shared__ int s_pins[BLOCK_NETS * MAX_DEG];   // 8 KB: block's contiguous CSR slice

    const int tid  = (int)threadIdx.x;
    const int net0 = (int)blockIdx.x * BLOCK_NETS;
    const int net  = net0 + tid;

    int net_end = net0 + BLOCK_NETS;
    if (net_end > n_nets) net_end = n_nets;

    // Block-uniform CSR range (scalar loads).
    const int s_blk = netpin_start[net0];
    int count = netpin_start[net_end] - s_blk;
    if (count > BLOCK_NETS * MAX_DEG) count = BLOCK_NETS * MAX_DEG;

#if USE_ASYNC_LDS
    for (int i = tid; i < count; i += BLOCK_NETS) {
        __builtin_amdgcn_global_load_async_to_lds_b32(
            (GInt*)(flat_netpin + s_blk + i),
            (LInt*)&s_pins[i],
            /*offset=*/0, /*cpol=*/0);
    }
    asm volatile("s_wait_asynccnt 0" ::: "memory");
#else
    for (int i = tid; i < count; i += BLOCK_NETS) {
        s_pins[i] = flat_netpin[s_blk + i];
    }
#endif
    __syncthreads();

    if (net >= n_nets) return;

    const int s   = netpin_start[net];
    const int deg = netpin_start[net + 1] - s;
    const int ignore_deg = ignore_net_degree_p[0];

    float result = 0.0f;
    if (deg >= 2 && deg <= ignore_deg) {
        const int degc = (deg < MAX_DEG) ? deg : MAX_DEG;
        const int base = s - s_blk;

        // Gather pin coordinates (clamp k>=deg to last pin, like the reference).
        float xs[MAX_DEG], ys[MAX_DEG];
        #pragma unroll
        for (int k = 0; k < MAX_DEG; ++k) {
            const int kk = (k < degc) ? k : (degc - 1);
            const int p  = s_pins[base + kk];
            xs[k] = pos[p];
            ys[k] = pos[num_pins + p];
        }

        // Prim: key[k] = min Manhattan distance from current tree to node k.
        float key[MAX_DEG];
        #pragma unroll
        for (int k = 1; k < MAX_DEG; ++k)
            key[k] = __builtin_fabsf(xs[k] - xs[0]) + __builtin_fabsf(ys[k] - ys[0]);

        unsigned avail = ((1u << degc) - 1u) & ~1u;   // nodes not yet in tree
        float total = 0.0f;

        #pragma unroll
        for (int it = 1; it < MAX_DEG; ++it) {
            // argmin over available keys; strict '<' => first-index tie-break
            // (matches jnp.argmin exactly).
            float c = INF_F;
            int jmin = 0;
            #pragma unroll
            for (int k = 1; k < MAX_DEG; ++k) {
                const float kv = ((avail >> k) & 1u) ? key[k] : INF_F;
                const bool  lt = kv < c;
                jmin = lt ? k : jmin;
                c    = lt ? kv : c;
            }
            // Only deg-1 steps actually contribute (identical gating to ref).
            total += (it < degc) ? c : 0.0f;
            avail &= ~(1u << jmin);

            // Select coords of new tree node via cndmask chain (no dyn indexing).
            float xj = xs[0], yj = ys[0];
            #pragma unroll
            for (int k = 1; k < MAX_DEG; ++k) {
                const bool sel = (jmin == k);
                xj = sel ? xs[k] : xj;
                yj = sel ? ys[k] : yj;
            }
            // Relax keys with distances to the new node (recompute, don't store 8x8).
            #pragma unroll
            for (int k = 1; k < MAX_DEG; ++k) {
                const float d = __builtin_fabsf(xs[k] - xj) + __builtin_fabsf(ys[k] - yj);
                key[k] = fminf(key[k], d);
            }
        }
        result = total;
    }
    out[net] = result;
}

extern "C" void kernel_launch(void* const* d_in, const int* in_sizes, int n_in,
                              void* d_out, int out_size, void* d_ws, size_t ws_size,
                              hipStream_t stream) {
    (void)n_in; (void)d_ws; (void)ws_size;
    const float* pos          = (const float*)d_in[0];
    const int*   flat_netpin  = (const int*)d_in[1];
    const int*   netpin_start = (const int*)d_in[2];
    const int*   ignore_deg   = (const int*)d_in[3];
    float*       out          = (float*)d_out;

    const int num_pins = in_sizes[1];
    const int n_nets   = in_sizes[2] - 1;   // == out_size

    const int block = BLOCK_NETS;
    const int grid  = (n_nets + block - 1) / block;
    rmst_wl_kernel<<<grid, block, 0, stream>>>(pos, flat_netpin, netpin_start,
                                               ignore_deg, out, n_nets, num_pins);
    (void)out_size;
}